// SecondGeometry_48601849921728
// MI455X (gfx1250) — compile-verified
//
#include <hip/hip_runtime.h>
#include <hip/hip_bf16.h>

typedef __attribute__((ext_vector_type(2))) float v2f;
typedef __attribute__((ext_vector_type(8))) float v8f;

#define BPTS   8192
#define DD     16
#define HH     128
#define THRESH 0.01f
#define LSTR   132   // LDS row stride (floats) for 16x128 staging: conflict-free b64 frag reads
#define RSTR   17    // LDS row stride for 16x16 reduction tiles
#define WPB    2     // waves per block

struct WaveLds {
  float h1  [16 * LSTR];  // h1 = tanh(P@W1+b1)            (staged, phases 2 & 4)
  float t   [16 * LSTR];  // da1 = V@W1, reused for t1=u2@W2^T
  float u2  [16 * LSTR];  // s2 .* w3
  float phiP[16 * RSTR];  // per-lane partial sums of phi
  float vP  [16 * RSTR];  // per-lane partial sums of v^T H v
  float gM  [16 * RSTR];  // g = grad phi (16x16)
};

__device__ __forceinline__ v8f wmma4(v2f a, v2f b, v8f c) {
  // D = A(16x4,f32) * B(4x16,f32) + C(16x16,f32)
  return __builtin_amdgcn_wmma_f32_16x16x4_f32(false, a, false, b, (short)0, c,
                                               false, false);
}

// gfx1250 has native v_tanh_f32 (TRANS32). Use it if the builtin is exposed;
// fall back to libm tanhf otherwise (branchy, ~30 VALU ops).
__device__ __forceinline__ float fast_tanh(float a) {
#if __has_builtin(__builtin_amdgcn_tanhf)
  return __builtin_amdgcn_tanhf(a);
#elif __has_builtin(__builtin_amdgcn_tanh_f32)
  return __builtin_amdgcn_tanh_f32(a);
#else
  return tanhf(a);
#endif
}

__global__ __launch_bounds__(32 * WPB)
void second_geometry_kernel(const float* __restrict__ x,
                            const float* __restrict__ attractor,
                            const float* __restrict__ k_s,
                            const float* __restrict__ k_d,
                            const float* __restrict__ W1,
                            const float* __restrict__ b1,
                            const float* __restrict__ W2,
                            const float* __restrict__ b2,
                            const float* __restrict__ w3,
                            const float* __restrict__ b3,
                            float* __restrict__ out) {
  __shared__ WaveLds lds[WPB];
  const int tid  = threadIdx.x;
  const int wv   = tid >> 5;
  const int lane = tid & 31;
  const int m16  = lane & 15;    // point row (A frag) or column index (B/C frag)
  const int half = lane >> 4;    // K/M half selector per ISA fragment layout
  const int koff = half * 2;
  WaveLds& L = lds[wv];

  const int tile = blockIdx.x * WPB + wv;
  const int pb   = tile * 16;    // 16 points per wave

  // Warm GL2 for the shared weight matrix (global_prefetch_b8 path).
#pragma unroll
  for (int i = 0; i < 8; ++i)
    __builtin_prefetch(W2 + (size_t)(tid + i * 64) * 32, 0, 1);

  // ---- Phase 0: A-fragments of P (pos) and V (vel), 16x16, K = point dim ----
  const float* xrow = x + (size_t)(pb + m16) * (2 * DD);
  v2f ap[4], av[4];
#pragma unroll
  for (int k4 = 0; k4 < 4; ++k4) {
    ap[k4] = *(const v2f*)(xrow + 4 * k4 + koff);
    av[k4] = *(const v2f*)(xrow + DD + 4 * k4 + koff);
  }

  // ---- Phase 1: a1 = P@W1 + b1 ; da1 = V@W1. Stage h1 = tanh(a1), da1. ----
#pragma unroll 1
  for (int nt = 0; nt < 8; ++nt) {
    const int col = nt * 16 + m16;
    v8f acc1 = {0.f,0.f,0.f,0.f,0.f,0.f,0.f,0.f};
    v8f accd = {0.f,0.f,0.f,0.f,0.f,0.f,0.f,0.f};
#pragma unroll
    for (int k4 = 0; k4 < 4; ++k4) {
      v2f bw;                                  // B = W1 (16x128), rows = K
      bw.x = W1[(4 * k4 + koff)     * HH + col];
      bw.y = W1[(4 * k4 + koff + 1) * HH + col];
      acc1 = wmma4(ap[k4], bw, acc1);
      accd = wmma4(av[k4], bw, accd);
    }
    const float b1v = b1[col];
#pragma unroll
    for (int i = 0; i < 8; ++i) {
      const int row = i + half * 8;            // C/D layout: row = i + 8*half
      L.h1[row * LSTR + col] = fast_tanh(acc1[i] + b1v);
      L.t [row * LSTR + col] = accd[i];
    }
  }

  // ---- Phase 2: {h1, h1', h1''} @ W2 with shared B fragments. ----
  // h1'=s1*da1, h1''=-2*h1*s1*da1^2 recomputed elementwise from staged frags.
  v8f phiAcc = {0.f,0.f,0.f,0.f,0.f,0.f,0.f,0.f};
  v8f vhvAcc = {0.f,0.f,0.f,0.f,0.f,0.f,0.f,0.f};
#pragma unroll 1
  for (int nt = 0; nt < 8; ++nt) {
    const int col = nt * 16 + m16;
    v8f A2   = {0.f,0.f,0.f,0.f,0.f,0.f,0.f,0.f};
    v8f A2p  = {0.f,0.f,0.f,0.f,0.f,0.f,0.f,0.f};
    v8f A2pp = {0.f,0.f,0.f,0.f,0.f,0.f,0.f,0.f};
#pragma unroll 8
    for (int ks = 0; ks < 32; ++ks) {
      const int kk = 4 * ks + koff;
      v2f fh = *(const v2f*)&L.h1[m16 * LSTR + kk];
      v2f fd = *(const v2f*)&L.t [m16 * LSTR + kk];
      v2f s1, fhp, fhpp;
      s1.x = 1.f - fh.x * fh.x;  s1.y = 1.f - fh.y * fh.y;
      fhp.x = s1.x * fd.x;       fhp.y = s1.y * fd.y;
      fhpp.x = -2.f * fh.x * s1.x * fd.x * fd.x;
      fhpp.y = -2.f * fh.y * s1.y * fd.y * fd.y;
      v2f bw;                                  // B = W2 (128x128)
      bw.x = W2[(size_t)(kk)     * HH + col];
      bw.y = W2[(size_t)(kk + 1) * HH + col];
      A2   = wmma4(fh,   bw, A2);
      A2p  = wmma4(fhp,  bw, A2p);
      A2pp = wmma4(fhpp, bw, A2pp);
    }
    const float b2v = b2[col];
    const float w3v = w3[col];
#pragma unroll
    for (int i = 0; i < 8; ++i) {
      const int row = i + half * 8;
      const float h2 = fast_tanh(A2[i] + b2v);
      const float s2 = 1.f - h2 * h2;
      phiAcc[i] += w3v * h2;
      const float a2p = A2p[i];
      vhvAcc[i] += w3v * (s2 * A2pp[i] - 2.f * h2 * s2 * a2p * a2p);
      L.u2[row * LSTR + col] = s2 * w3v;
    }
  }
#pragma unroll
  for (int i = 0; i < 8; ++i) {
    const int row = i + half * 8;
    L.phiP[row * RSTR + m16] = phiAcc[i];
    L.vP [row * RSTR + m16]  = vhvAcc[i];
  }

  // ---- Phase 3: t1 = u2 @ W2^T  (B frag = contiguous W2 row pair) ----
#pragma unroll 1
  for (int nt = 0; nt < 8; ++nt) {
    const int n = nt * 16 + m16;
    v8f T = {0.f,0.f,0.f,0.f,0.f,0.f,0.f,0.f};
#pragma unroll 8
    for (int ks = 0; ks < 32; ++ks) {
      const int kk = 4 * ks + koff;
      v2f fu = *(const v2f*)&L.u2[m16 * LSTR + kk];
      v2f bw = *(const v2f*)(W2 + (size_t)n * HH + kk);  // (W2^T)[k][n] = W2[n][k]
      T = wmma4(fu, bw, T);
    }
#pragma unroll
    for (int i = 0; i < 8; ++i)
      L.t[(i + half * 8) * LSTR + n] = T[i];
  }

  // ---- Phase 4: g = (s1 .* t1) @ W1^T  → 16x16 ----
  v8f G = {0.f,0.f,0.f,0.f,0.f,0.f,0.f,0.f};
#pragma unroll 8
  for (int ks = 0; ks < 32; ++ks) {
    const int kk = 4 * ks + koff;
    v2f fh = *(const v2f*)&L.h1[m16 * LSTR + kk];
    v2f ft = *(const v2f*)&L.t [m16 * LSTR + kk];
    v2f fu1;
    fu1.x = (1.f - fh.x * fh.x) * ft.x;
    fu1.y = (1.f - fh.y * fh.y) * ft.y;
    v2f bw = *(const v2f*)(W1 + (size_t)m16 * HH + kk); // (W1^T)[k][d] = W1[d][k]
    G = wmma4(fu1, bw, G);
  }
#pragma unroll
  for (int i = 0; i < 8; ++i)
    L.gM[(i + half * 8) * RSTR + m16] = G[i];

  // ---- Phase 5: closed-form epilogue, one lane per point ----
  const float ksc = k_s[0];
  const float kdc = k_d[0];
  const float b3v = b3[0];
  if (half == 0) {
    const int m = m16;
    const float* xr = x + (size_t)(pb + m) * (2 * DD);
    float phi = b3v, vhv = 0.f;
#pragma unroll
    for (int j = 0; j < 16; ++j) {
      phi += L.phiP[m * RSTR + j];
      vhv += L.vP [m * RSTR + j];
    }
    const float hw = (fabsf(phi) < THRESH) ? 1.f : 0.f;
    float s = 0.f, gf = 0.f;
    float gv[16], fv[16];
#pragma unroll
    for (int d = 0; d < 16; ++d) {
      const float gd = L.gM[m * RSTR + d];
      const float f  = -hw * (kdc * xr[DD + d] + ksc * (xr[d] - attractor[d]));
      gv[d] = gd; fv[d] = f;
      s  += gd * gd;
      gf += gd * f;
    }
    const float coef = (gf + vhv) / (1.f + s);
    float* orow = out + (size_t)(pb + m) * DD;
#pragma unroll
    for (int d = 0; d < 16; ++d)
      orow[d] = fv[d] - gv[d] * coef;
  }
}

extern "C" void kernel_launch(void* const* d_in, const int* in_sizes, int n_in,
                              void* d_out, int out_size, void* d_ws, size_t ws_size,
                              hipStream_t stream) {
  const float* x         = (const float*)d_in[0];
  const float* attractor = (const float*)d_in[1];
  const float* k_s       = (const float*)d_in[2];
  const float* k_d       = (const float*)d_in[3];
  const float* W1        = (const float*)d_in[4];
  const float* b1        = (const float*)d_in[5];
  const float* W2        = (const float*)d_in[6];
  const float* b2        = (const float*)d_in[7];
  const float* w3        = (const float*)d_in[8];
  const float* b3        = (const float*)d_in[9];
  float* out = (float*)d_out;

  dim3 grid(BPTS / 16 / WPB);   // 256 blocks, 2 waves each, 16 points per wave
  dim3 block(32 * WPB);
  hipLaunchKernelGGL(second_geometry_kernel, grid, block, 0, stream,
                     x, attractor, k_s, k_d, W1, b1, W2, b2, w3, b3, out);
  (void)in_sizes; (void)n_in; (void)out_size; (void)d_ws; (void)ws_size;
}